// GATE_70970039599164
// MI455X (gfx1250) — compile-verified
//
#include <hip/hip_runtime.h>
#include <hip/hip_bf16.h>

// ---------------------------------------------------------------------------
// Problem constants (match reference)
// ---------------------------------------------------------------------------
#define NN     20000      // nodes
#define EE     320000     // edges
#define EP     (EE + NN)  // edges + self loops
#define HH     4          // heads
#define CC     128        // per-head channels
#define HC     (HH * CC)  // 512
#define INF_   64         // in_channels
#define EDD    16         // edge_dim
#define GG     128        // graphs
#define OUTC   128        // out_channels
#define C3     (3 * CC)   // 384
#define NEG_SLOPE 0.2f

typedef float v2f __attribute__((ext_vector_type(2)));
typedef float v8f __attribute__((ext_vector_type(8)));

// monotone float <-> uint order map (for atomicMax on floats)
__device__ __forceinline__ unsigned f2ord(float f) {
    unsigned u = __float_as_uint(f);
    return (u >> 31) ? ~u : (u | 0x80000000u);
}
__device__ __forceinline__ float ord2f(unsigned u) {
    return __uint_as_float((u >> 31) ? (u & 0x7FFFFFFFu) : ~u);
}

// ---------------------------------------------------------------------------
// Generic fp32 WMMA GEMM: C[M,Ncol] = A[M,K] @ B[K,Ncol] (+bias, +relu)
// M % 16 == 0, Ncol % 16 == 0, K % 4 == 0. One 16x16 tile per wave.
// Uses V_WMMA_F32_16X16X4_F32 accumulating over K in steps of 4.
// ---------------------------------------------------------------------------
__global__ void k_gemm_wmma_f32(const float* __restrict__ A, int lda,
                                const float* __restrict__ B, int ldb,
                                float* __restrict__ Cout, int ldc,
                                int M, int K, int Ncol,
                                const float* __restrict__ bias, int relu)
{
    const int lane = threadIdx.x & 31;
    const int wave = threadIdx.x >> 5;
    const int tile = blockIdx.x * (blockDim.x >> 5) + wave;
    const int tilesN = Ncol >> 4;
    const int numTiles = (M >> 4) * tilesN;
    if (tile >= numTiles) return;          // whole-wave uniform exit: EXEC stays all-1 for WMMA

    const int tm = tile / tilesN;
    const int tn = tile % tilesN;
    const int hi = lane >> 4;              // 0: lanes 0-15, 1: lanes 16-31
    const int lo = lane & 15;

    const float* Arow = A + (size_t)(tm * 16 + lo) * lda;   // A: 16x4 tile, row = lo
    const float* Bcol = B + tn * 16 + lo;                   // B: 4x16 tile, col = lo

    v8f c = {};
    for (int k = 0; k < K; k += 4) {
        v2f a, b;
        // A 16x4 layout: v0 holds K=(2*hi), v1 holds K=(2*hi+1), rows striped over lanes 0-15
        a.x = Arow[k + 2 * hi];
        a.y = Arow[k + 2 * hi + 1];
        // B 4x16 layout: symmetric — rows K=(2*hi), K=(2*hi+1) striped across lane columns
        b.x = Bcol[(size_t)(k + 2 * hi) * ldb];
        b.y = Bcol[(size_t)(k + 2 * hi + 1) * ldb];
        c = __builtin_amdgcn_wmma_f32_16x16x4_f32(false, a, false, b,
                                                  (short)0, c, false, false);
    }

    const int col = tn * 16 + lo;
    const float bv = bias ? bias[col] : 0.0f;
#pragma unroll
    for (int r = 0; r < 8; ++r) {
        // C/D layout: VGPR r holds M = r (lanes 0-15) and M = r+8 (lanes 16-31)
        const int row = tm * 16 + r + 8 * hi;
        float v = c[r] + bv;
        if (relu) v = v > 0.0f ? v : 0.0f;
        Cout[(size_t)row * ldc + col] = v;
    }
}

// ---------------------------------------------------------------------------
// Edge-attr column sums (for self-loop fill_value='mean')
// ---------------------------------------------------------------------------
__global__ void k_edge_mean(const float* __restrict__ ea, float* __restrict__ msum)
{
    __shared__ float s[EDD];
    if (threadIdx.x < EDD) s[threadIdx.x] = 0.0f;
    __syncthreads();
    for (int e = blockIdx.x * blockDim.x + threadIdx.x; e < EE;
         e += gridDim.x * blockDim.x) {
        const float* r = ea + (size_t)e * EDD;
#pragma unroll
        for (int d = 0; d < EDD; ++d) atomicAdd(&s[d], r[d]);
    }
    __syncthreads();
    if (threadIdx.x < EDD) atomicAdd(&msum[threadIdx.x], s[threadIdx.x]);
}

__global__ void k_edge_mean_finish(float* msum)
{
    if (threadIdx.x < EDD) msum[threadIdx.x] *= (1.0f / (float)EE);
}

// ---------------------------------------------------------------------------
// Ve[d,h] = sum_c We[d, h*C + c] * a_edge[h, c]   (collapses ep-GEMM)
// ---------------------------------------------------------------------------
__global__ void k_compute_ve(const float* __restrict__ We,
                             const float* __restrict__ ae,
                             float* __restrict__ Ve)
{
    const int t = threadIdx.x;           // 64 threads
    const int d = t >> 2, h = t & 3;
    float s = 0.0f;
    for (int c = 0; c < CC; ++c)
        s += We[(size_t)d * HC + h * CC + c] * ae[h * CC + c];
    Ve[d * HH + h] = s;
}

// ---------------------------------------------------------------------------
// Per-(node, head) attention dots: al_s = <xs[n,h,:], a_src[h]>, al_d likewise
// One wave32 per (n,h); lane-parallel dot + shfl_xor reduction.
// ---------------------------------------------------------------------------
__global__ void k_att_node(const float* __restrict__ xs,
                           const float* __restrict__ as_,
                           const float* __restrict__ ad_,
                           float* __restrict__ als, float* __restrict__ ald)
{
    const int w = (blockIdx.x * blockDim.x + threadIdx.x) >> 5;
    const int lane = threadIdx.x & 31;
    if (w >= NN * HH) return;
    const int n = w >> 2, h = w & 3;
    const float* xr  = xs  + (size_t)n * HC + h * CC;
    const float* asr = as_ + h * CC;
    const float* adr = ad_ + h * CC;
    float s = 0.0f, d = 0.0f;
    for (int c = lane; c < CC; c += 32) {
        const float v = xr[c];
        s += v * asr[c];
        d += v * adr[c];
    }
    for (int off = 16; off; off >>= 1) {
        s += __shfl_xor(s, off, 32);
        d += __shfl_xor(d, off, 32);
    }
    if (lane == 0) { als[w] = s; ald[w] = d; }
}

// ---------------------------------------------------------------------------
// alpha[e,h] = leaky_relu(al_s[src] + al_d[dst] + eattr[e].Ve[:,h]); seg-max
// ---------------------------------------------------------------------------
__global__ void k_alpha(const int* __restrict__ srcA, const int* __restrict__ dstA,
                        const float* __restrict__ ea, const float* __restrict__ mean_e,
                        const float* __restrict__ Ve,
                        const float* __restrict__ als, const float* __restrict__ ald,
                        float* __restrict__ alpha, unsigned* __restrict__ dmax)
{
    const int e = blockIdx.x * blockDim.x + threadIdx.x;
    if (e >= EP) return;
    int s, d; const float* row;
    if (e < EE) { s = srcA[e]; d = dstA[e]; row = ea + (size_t)e * EDD; }
    else        { s = d = e - EE;           row = mean_e; }
    float r[EDD];
#pragma unroll
    for (int i = 0; i < EDD; ++i) r[i] = row[i];
#pragma unroll
    for (int h = 0; h < HH; ++h) {
        float ale = 0.0f;
#pragma unroll
        for (int i = 0; i < EDD; ++i) ale += r[i] * Ve[i * HH + h];
        float a = als[s * HH + h] + ald[d * HH + h] + ale;
        a = a > 0.0f ? a : NEG_SLOPE * a;
        alpha[(size_t)e * HH + h] = a;
        atomicMax(&dmax[d * HH + h], f2ord(a));
    }
}

// ---------------------------------------------------------------------------
// ex = exp(alpha - max[dst]); denom[dst] += ex   (alpha overwritten with ex)
// ---------------------------------------------------------------------------
__global__ void k_exp(const int* __restrict__ dstA, const unsigned* __restrict__ dmax,
                      float* __restrict__ alpha, float* __restrict__ denom)
{
    const int idx = blockIdx.x * blockDim.x + threadIdx.x;
    if (idx >= EP * HH) return;
    const int e = idx >> 2, h = idx & 3;
    const int d = (e < EE) ? dstA[e] : (e - EE);
    const float m = ord2f(dmax[d * HH + h]);
    const float ex = __expf(alpha[idx] - m);
    alpha[idx] = ex;
    atomicAdd(&denom[d * HH + h], ex);
}

// ---------------------------------------------------------------------------
// Message scatter: agg[dst, h, c] += (ex/denom) * xs[src, h, c]
// One wave per edge; 32 lanes cover H*C=512 in 16 strided steps.
// ---------------------------------------------------------------------------
__global__ void k_scatter(const int* __restrict__ srcA, const int* __restrict__ dstA,
                          const float* __restrict__ xs, const float* __restrict__ alpha,
                          const float* __restrict__ denom, float* __restrict__ agg)
{
    const int e = (blockIdx.x * blockDim.x + threadIdx.x) >> 5;
    const int lane = threadIdx.x & 31;
    if (e >= EP) return;
    int s, d;
    if (e < EE) { s = srcA[e]; d = dstA[e]; } else { s = d = e - EE; }
    const float* xrow = xs  + (size_t)s * HC;
    float*       arow = agg + (size_t)d * HC;
    for (int i = lane; i < HC; i += 32) {
        const int h = i >> 7;
        const float coeff = alpha[(size_t)e * HH + h] / denom[d * HH + h];
        atomicAdd(&arow[i], coeff * xrow[i]);
    }
}

// ---------------------------------------------------------------------------
// Head mean + bias -> h_out; simultaneously pool into pooled[g, layer*C + c]
// ---------------------------------------------------------------------------
__global__ void k_mean_bias_pool(const float* __restrict__ agg,
                                 const float* __restrict__ bias,
                                 const int* __restrict__ batch,
                                 float* __restrict__ hout,
                                 float* __restrict__ pooled, int layer)
{
    const int idx = blockIdx.x * blockDim.x + threadIdx.x;
    if (idx >= NN * CC) return;
    const int n = idx >> 7, c = idx & 127;
    const float* ar = agg + (size_t)n * HC;
    const float v = 0.25f * (ar[c] + ar[CC + c] + ar[2 * CC + c] + ar[3 * CC + c])
                    + bias[c];
    hout[idx] = v;
    atomicAdd(&pooled[(size_t)batch[n] * C3 + layer * CC + c], v);
}

// ---------------------------------------------------------------------------
// Host-side launcher
// ---------------------------------------------------------------------------
extern "C" void kernel_launch(void* const* d_in, const int* in_sizes, int n_in,
                              void* d_out, int out_size, void* d_ws, size_t ws_size,
                              hipStream_t stream)
{
    (void)in_sizes; (void)n_in; (void)out_size; (void)ws_size;

    const float* x   = (const float*)d_in[0];
    const int*   ei  = (const int*)d_in[1];     // [2,E] row-major: src then dst
    const float* ea  = (const float*)d_in[2];
    const int*   bat = (const int*)d_in[3];
    const int* srcA = ei;
    const int* dstA = ei + EE;

    const float* fc1_w = (const float*)d_in[22];
    const float* fc1_b = (const float*)d_in[23];
    const float* fc2_w = (const float*)d_in[24];
    const float* fc2_b = (const float*)d_in[25];
    float* out = (float*)d_out;

    // workspace layout (floats)
    float* ws = (float*)d_ws;
    size_t o = 0;
    float*    xs_    = ws + o; o += (size_t)NN * HC;   // projected features
    float*    agg    = ws + o; o += (size_t)NN * HC;   // message accumulator
    float*    hbuf   = ws + o; o += (size_t)NN * CC;   // layer output
    float*    alpha  = ws + o; o += (size_t)EP * HH;   // attention logits -> exp
    float*    als    = ws + o; o += (size_t)NN * HH;
    float*    ald    = ws + o; o += (size_t)NN * HH;
    unsigned* dmax   = (unsigned*)(ws + o); o += (size_t)NN * HH;
    float*    denom  = ws + o; o += (size_t)NN * HH;
    float*    pooled = ws + o; o += (size_t)GG * C3;
    float*    hidden = ws + o; o += (size_t)GG * C3;
    float*    Ve     = ws + o; o += (size_t)EDD * HH;
    float*    mean_e = ws + o; o += (size_t)EDD;

    // one-time zeroing
    hipMemsetAsync(mean_e, 0, EDD * sizeof(float), stream);
    hipMemsetAsync(pooled, 0, (size_t)GG * C3 * sizeof(float), stream);

    // mean edge attribute (self-loop fill)
    k_edge_mean<<<512, 256, 0, stream>>>(ea, mean_e);
    k_edge_mean_finish<<<1, 16, 0, stream>>>(mean_e);

    const int gemmTilesNode = (NN / 16) * (HC / 16);          // 1250 * 32
    const int gemmBlocksNode = (gemmTilesNode + 7) / 8;       // 8 waves / block

    for (int l = 0; l < 3; ++l) {
        const float* W   = (const float*)d_in[4 + 6 * l];
        const float* We  = (const float*)d_in[5 + 6 * l];
        const float* as_ = (const float*)d_in[6 + 6 * l];
        const float* ad_ = (const float*)d_in[7 + 6 * l];
        const float* ae_ = (const float*)d_in[8 + 6 * l];
        const float* b_  = (const float*)d_in[9 + 6 * l];
        const float* xin = (l == 0) ? x : hbuf;
        const int K = (l == 0) ? INF_ : CC;

        k_compute_ve<<<1, 64, 0, stream>>>(We, ae_, Ve);

        // xs = xin @ W   [N, HC]
        k_gemm_wmma_f32<<<gemmBlocksNode, 256, 0, stream>>>(
            xin, K, W, HC, xs_, HC, NN, K, HC, nullptr, 0);

        // attention node dots
        k_att_node<<<(NN * HH + 7) / 8, 256, 0, stream>>>(xs_, as_, ad_, als, ald);

        // zero per-layer accumulators
        hipMemsetAsync(agg,   0, (size_t)NN * HC * sizeof(float), stream);
        hipMemsetAsync(dmax,  0, (size_t)NN * HH * sizeof(unsigned), stream);
        hipMemsetAsync(denom, 0, (size_t)NN * HH * sizeof(float), stream);

        k_alpha<<<(EP + 255) / 256, 256, 0, stream>>>(
            srcA, dstA, ea, mean_e, Ve, als, ald, alpha, dmax);

        k_exp<<<(EP * HH + 255) / 256, 256, 0, stream>>>(dstA, dmax, alpha, denom);

        k_scatter<<<(EP * 32 + 255) / 256, 256, 0, stream>>>(
            srcA, dstA, xs_, alpha, denom, agg);

        k_mean_bias_pool<<<(NN * CC + 255) / 256, 256, 0, stream>>>(
            agg, b_, bat, hbuf, pooled, l);
    }

    // MLP head: relu(pooled @ fc1_w + fc1_b) @ fc2_w + fc2_b
    const int t1 = (GG / 16) * (C3 / 16);     // 8 * 24
    k_gemm_wmma_f32<<<(t1 + 7) / 8, 256, 0, stream>>>(
        pooled, C3, fc1_w, C3, hidden, C3, GG, C3, C3, fc1_b, 1);

    const int t2 = (GG / 16) * (OUTC / 16);   // 8 * 8
    k_gemm_wmma_f32<<<(t2 + 7) / 8, 256, 0, stream>>>(
        hidden, C3, fc2_w, OUTC, out, OUTC, GG, C3, OUTC, fc2_b, 0);
}